// DAGCN_89472758710366
// MI455X (gfx1250) — compile-verified
//
#include <hip/hip_runtime.h>
#include <hip/hip_bf16.h>

#define BATCH 4096
#define CHAN  62
#define NHEAD 5

typedef __attribute__((ext_vector_type(2))) float v2f;
typedef __attribute__((ext_vector_type(8))) float v8f;

__device__ __forceinline__ v8f wmma4(v2f a, v2f b, v8f c) {
  // D = A(16x4,f32) * B(4x16,f32) + C(16x16,f32) ; 8-arg form:
  // (neg_a, A, neg_b, B, c_mod, C, reuse_a, reuse_b)
  return __builtin_amdgcn_wmma_f32_16x16x4_f32(false, a, false, b, (short)0, c,
                                               false, false);
}

__device__ __forceinline__ float eluf(float v) {
  return v > 0.f ? v : expm1f(v);
}

// -------- workspace layout (float offsets) --------
#define OFF_ADS   0u                  // 3844
#define OFF_L     3844u               // 3844
#define OFF_FIN1  7688u               // 20  (scale,shift per br,band) conv1-bn
#define OFF_FIN2  7708u               // 20  conv2-bn
#define OFF_P1    7728u               // 2*4096*5*2 = 81920 partial stats conv1
#define OFF_P2    89648u              // 81920 partial stats conv2
#define OFF_H1    171568u             // 2*4096*5*60 = 2457600
#define OFF_H2    2629168u            // 2*4096*5*62 = 2539520
#define OFF_F12   5168688u            // 2*4096*62*5 = 2539520 (gcn out, per branch)
#define OFF_CAT   7708208u            // 4096*62*10 = 2539520 (concat / enc0 buffer)

// ============================================================
// A_ds = relu(tanh(elu(A@w1)@w2)); L = A_ds / colsum(A_ds)
// ============================================================
__global__ __launch_bounds__(256)
void k_prep(const float* __restrict__ A, const float* __restrict__ w1,
            const float* __restrict__ w2, float* __restrict__ Ads,
            float* __restrict__ Lm)
{
  __shared__ float red[256];
  __shared__ float hb[30];
  __shared__ float sh[3844];
  __shared__ float colsum[62];
  const int tid = threadIdx.x;

  for (int j = 0; j < 30; ++j) {
    float s = 0.f;
    for (int i = tid; i < 3844; i += 256) s += A[i] * w1[i * 30 + j];
    red[tid] = s; __syncthreads();
    for (int o = 128; o > 0; o >>= 1) {
      if (tid < o) red[tid] += red[tid + o];
      __syncthreads();
    }
    if (tid == 0) hb[j] = eluf(red[0]);
    __syncthreads();
  }
  for (int k = tid; k < 3844; k += 256) {
    float s = 0.f;
    for (int j = 0; j < 30; ++j) s += hb[j] * w2[j * 3844 + k];
    float t = tanhf(s);
    sh[k] = t > 0.f ? t : 0.f;
  }
  __syncthreads();
  for (int p = tid; p < 62; p += 256) {
    float s = 0.f;
    for (int i = 0; i < 62; ++i) s += sh[i * 62 + p];
    colsum[p] = s;
  }
  __syncthreads();
  for (int k = tid; k < 3844; k += 256) {
    Ads[k] = sh[k];
    Lm[k]  = sh[k] / colsum[k % 62];
  }
}

// ============================================================
// depthwise conv1 (k=3, no pad) over chan axis; per-block stat partials
// ============================================================
__global__ __launch_bounds__(128)
void k_conv1(const float* __restrict__ x, const float* __restrict__ k1a,
             const float* __restrict__ k1b, float* __restrict__ h1,
             float* __restrict__ part)
{
  const int b = blockIdx.x, br = blockIdx.y, tid = threadIdx.x;
  const float* k1 = br ? k1b : k1a;
  __shared__ float xt[5 * 62];
  __shared__ float red[128];
  for (int i = tid; i < 310; i += 128) {
    int band = i / 62, c = i % 62;
    xt[i] = x[(b * 62 + c) * 10 + br * 5 + band];
  }
  __syncthreads();
  float s[5] = {0, 0, 0, 0, 0}, s2[5] = {0, 0, 0, 0, 0};
  for (int i = tid; i < 300; i += 128) {
    int band = i / 60, w = i % 60;
    float v = xt[band * 62 + w] * k1[band * 3] +
              xt[band * 62 + w + 1] * k1[band * 3 + 1] +
              xt[band * 62 + w + 2] * k1[band * 3 + 2];
    h1[((size_t)(br * BATCH + b) * 5 + band) * 60 + w] = v;
    s[band] += v; s2[band] += v * v;
  }
  for (int band = 0; band < 5; ++band) {
    for (int st = 0; st < 2; ++st) {
      red[tid] = st ? s2[band] : s[band];
      __syncthreads();
      for (int o = 64; o > 0; o >>= 1) {
        if (tid < o) red[tid] += red[tid + o];
        __syncthreads();
      }
      if (tid == 0)
        part[((size_t)(br * BATCH + b) * 5 + band) * 2 + st] = red[0];
      __syncthreads();
    }
  }
}

// ============================================================
// finalize bn stats (deterministic fixed-order reduction)
// fin[br][band] = (scale, shift)
// ============================================================
__global__ __launch_bounds__(256)
void k_finstats(const float* __restrict__ part, float* __restrict__ fin,
                const float* __restrict__ gA, const float* __restrict__ bA,
                const float* __restrict__ gB, const float* __restrict__ bB,
                float Ninv)
{
  __shared__ float red[256];
  __shared__ float sums[20];
  const int tid = threadIdx.x;
  for (int c = 0; c < 20; ++c) {
    int br = c / 10, band = (c % 10) / 2, st = c & 1;
    float s = 0.f;
    for (int b = tid; b < BATCH; b += 256)
      s += part[((size_t)(br * BATCH + b) * 5 + band) * 2 + st];
    red[tid] = s; __syncthreads();
    for (int o = 128; o > 0; o >>= 1) {
      if (tid < o) red[tid] += red[tid + o];
      __syncthreads();
    }
    if (tid == 0) sums[c] = red[0];
    __syncthreads();
  }
  if (tid < 10) {
    int br = tid / 5, band = tid % 5;
    float mean = sums[br * 10 + band * 2] * Ninv;
    float var  = sums[br * 10 + band * 2 + 1] * Ninv - mean * mean;
    float inv  = rsqrtf(var + 1e-5f);
    const float* g = br ? gB : gA;
    const float* bb = br ? bB : bA;
    float scale = g[band] * inv;
    fin[(br * 5 + band) * 2 + 0] = scale;
    fin[(br * 5 + band) * 2 + 1] = bb[band] - mean * scale;
  }
}

// ============================================================
// elu(bn1(h1)) -> conv2 (k=1, pad 1 each side -> width 62) ; stat partials
// ============================================================
__global__ __launch_bounds__(128)
void k_conv2(const float* __restrict__ h1, const float* __restrict__ fin1,
             const float* __restrict__ k2a, const float* __restrict__ k2b,
             float* __restrict__ h2, float* __restrict__ part)
{
  const int b = blockIdx.x, br = blockIdx.y, tid = threadIdx.x;
  const float* k2 = br ? k2b : k2a;
  __shared__ float red[128];
  float s[5] = {0, 0, 0, 0, 0}, s2[5] = {0, 0, 0, 0, 0};
  for (int i = tid; i < 310; i += 128) {
    int band = i / 62, w = i % 62;
    float v = 0.f;
    if (w >= 1 && w <= 60) {
      float hv = h1[((size_t)(br * BATCH + b) * 5 + band) * 60 + (w - 1)];
      float bn = hv * fin1[(br * 5 + band) * 2] + fin1[(br * 5 + band) * 2 + 1];
      v = eluf(bn) * k2[band];
    }
    h2[((size_t)(br * BATCH + b) * 5 + band) * 62 + w] = v;
    s[band] += v; s2[band] += v * v;
  }
  for (int band = 0; band < 5; ++band) {
    for (int st = 0; st < 2; ++st) {
      red[tid] = st ? s2[band] : s[band];
      __syncthreads();
      for (int o = 64; o > 0; o >>= 1) {
        if (tid < o) red[tid] += red[tid + o];
        __syncthreads();
      }
      if (tid == 0)
        part[((size_t)(br * BATCH + b) * 5 + band) * 2 + st] = red[0];
      __syncthreads();
    }
  }
}

// ============================================================
// y = elu( bn2(h2) @ L + xt ) ; write feat[br][b][chan][band]
// ============================================================
__global__ __launch_bounds__(128)
void k_gcn_out(const float* __restrict__ x, const float* __restrict__ h2,
               const float* __restrict__ fin2, const float* __restrict__ Lm,
               float* __restrict__ feat)
{
  const int b = blockIdx.x, br = blockIdx.y, tid = threadIdx.x;
  __shared__ float hb[5 * 62];
  __shared__ float xt[5 * 62];
  for (int i = tid; i < 310; i += 128) {
    int band = i / 62, c = i % 62;
    float v = h2[((size_t)(br * BATCH + b) * 5 + band) * 62 + c];
    hb[i] = v * fin2[(br * 5 + band) * 2] + fin2[(br * 5 + band) * 2 + 1];
    xt[i] = x[(b * 62 + c) * 10 + br * 5 + band];
  }
  __syncthreads();
  for (int i = tid; i < 310; i += 128) {
    int band = i / 62, p = i % 62;
    float s = 0.f;
    for (int k = 0; k < 62; ++k) s += hb[band * 62 + k] * Lm[k * 62 + p];
    feat[((size_t)(br * BATCH + b) * 62 + p) * 5 + band] = eluf(s + xt[band * 62 + p]);
  }
}

// ============================================================
// Fused 2-layer transformer encoder, one sample per block, WMMA f32 16x16x4
// ============================================================
struct LayerP {
  const float *wq, *wk, *wv, *fc, *aw1, *ab1, *aw2, *ab2;
  const float *f1, *f2, *fw1, *fb1, *fw2, *fb2;
};
struct EncP { LayerP L[2]; };

template <int DM>
__global__ __launch_bounds__(128)
void k_encoder(EncP P, const float* __restrict__ in, int in_ld, int in_off,
               float* __restrict__ out, int out_ld, int out_off,
               const float* __restrict__ Ads)
{
  __shared__ float Xp[64 * 16];          // x, zero padded (rows>=62 & cols>=DM = 0)
  __shared__ float Qm[64 * 40];
  __shared__ float Km[64 * 40];
  __shared__ float Vm[64 * 40];
  __shared__ float SC[64 * 64];          // scores / attn
  __shared__ float CTh[64 * 8];          // per-head context
  __shared__ float O1[64 * 16];          // fc output

  const int tid = threadIdx.x;
  const int wave = tid >> 5, lane = tid & 31;
  const int lh = lane & 15, hi = lane >> 4;
  const int b = blockIdx.x;
  const int mb = wave * 16;              // this wave's M-tile

  for (int i = tid; i < 64 * 16; i += 128) Xp[i] = 0.f;
  __syncthreads();
  for (int i = tid; i < 62 * DM; i += 128) {
    int c = i / DM, d = i % DM;
    Xp[c * 16 + d] = in[(size_t)(b * 62 + c) * in_ld + in_off + d];
  }
  __syncthreads();

  for (int layer = 0; layer < 2; ++layer) {
    const LayerP L = P.L[layer];

    // ---------- Q/K/V: Xp(64x16) @ W(DMx40, zero padded) ----------
    {
      const float* Ws[3] = {L.wq, L.wk, L.wv};
      float* Ds[3] = {Qm, Km, Vm};
      for (int w = 0; w < 3; ++w) {
        const float* W = Ws[w];
        float* D = Ds[w];
        for (int nt = 0; nt < 3; ++nt) {
          int nb = nt * 16, n = nb + lh;
          v8f acc = {0.f, 0.f, 0.f, 0.f, 0.f, 0.f, 0.f, 0.f};
          for (int ks = 0; ks < 4; ++ks) {
            int k0 = ks * 4 + hi * 2;
            v2f a, bb;
            a.x = Xp[(mb + lh) * 16 + k0];
            a.y = Xp[(mb + lh) * 16 + k0 + 1];
            bb.x = (k0 < DM && n < 40) ? W[k0 * 40 + n] : 0.f;
            bb.y = (k0 + 1 < DM && n < 40) ? W[(k0 + 1) * 40 + n] : 0.f;
            acc = wmma4(a, bb, acc);
          }
          for (int r = 0; r < 8; ++r)
            if (n < 40) D[(mb + r + hi * 8) * 40 + n] = acc[r];
        }
      }
    }
    __syncthreads();

    // ---------- attention heads; accumulate fc output in registers ----------
    v8f oacc = {0.f, 0.f, 0.f, 0.f, 0.f, 0.f, 0.f, 0.f};
    for (int h = 0; h < NHEAD; ++h) {
      // scores = Q_h(64x8) @ K_h^T(8x64)
      for (int nt = 0; nt < 4; ++nt) {
        int nb = nt * 16;
        v8f acc = {0.f, 0.f, 0.f, 0.f, 0.f, 0.f, 0.f, 0.f};
        for (int ks = 0; ks < 2; ++ks) {
          int k0 = ks * 4 + hi * 2;
          v2f a, bb;
          a.x = Qm[(mb + lh) * 40 + h * 8 + k0];
          a.y = Qm[(mb + lh) * 40 + h * 8 + k0 + 1];
          bb.x = Km[(nb + lh) * 40 + h * 8 + k0];
          bb.y = Km[(nb + lh) * 40 + h * 8 + k0 + 1];
          acc = wmma4(a, bb, acc);
        }
        for (int r = 0; r < 8; ++r)
          SC[(mb + r + hi * 8) * 64 + nb + lh] = acc[r];
      }
      __syncthreads();

      // softmax over k<62, scale 1/sqrt(8), then * A_ds[q][k]; pad cols -> 0
      if (tid < 64) {
        int q = tid;
        if (q < 62) {
          const float sc = 0.35355339059327373f;
          float mx = -1e30f;
          for (int k = 0; k < 62; ++k) mx = fmaxf(mx, SC[q * 64 + k] * sc);
          float sum = 0.f;
          for (int k = 0; k < 62; ++k) {
            float e = expf(SC[q * 64 + k] * sc - mx);
            SC[q * 64 + k] = e; sum += e;
          }
          float inv = 1.f / sum;
          for (int k = 0; k < 62; ++k)
            SC[q * 64 + k] = SC[q * 64 + k] * inv * Ads[q * 62 + k];
          SC[q * 64 + 62] = 0.f; SC[q * 64 + 63] = 0.f;
        } else {
          for (int k = 0; k < 64; ++k) SC[q * 64 + k] = 0.f;
        }
      }
      __syncthreads();

      // ctx_h = attn(64x64) @ V_h(64x8)  -> CTh (wave-local rows)
      {
        v8f acc = {0.f, 0.f, 0.f, 0.f, 0.f, 0.f, 0.f, 0.f};
        for (int ks = 0; ks < 16; ++ks) {
          int k0 = ks * 4 + hi * 2;
          v2f a, bb;
          a.x = SC[(mb + lh) * 64 + k0];
          a.y = SC[(mb + lh) * 64 + k0 + 1];
          bb.x = (lh < 8) ? Vm[k0 * 40 + h * 8 + lh] : 0.f;
          bb.y = (lh < 8) ? Vm[(k0 + 1) * 40 + h * 8 + lh] : 0.f;
          acc = wmma4(a, bb, acc);
        }
        for (int r = 0; r < 8; ++r)
          if (lh < 8) CTh[(mb + r + hi * 8) * 8 + lh] = acc[r];
      }
      // fc partial: CTh(64x8) @ fc[h*8 .. h*8+8, 0:DM]  (accumulate)
      for (int ks = 0; ks < 2; ++ks) {
        int k0 = ks * 4 + hi * 2;
        v2f a, bb;
        a.x = CTh[(mb + lh) * 8 + k0];
        a.y = CTh[(mb + lh) * 8 + k0 + 1];
        bb.x = (lh < DM) ? L.fc[(h * 8 + k0) * DM + lh] : 0.f;
        bb.y = (lh < DM) ? L.fc[(h * 8 + k0 + 1) * DM + lh] : 0.f;
        oacc = wmma4(a, bb, oacc);
      }
      __syncthreads();   // SC reused next head
    }
    for (int r = 0; r < 8; ++r) O1[(mb + r + hi * 8) * 16 + lh] = oacc[r];
    __syncthreads();

    // ---------- adapter + residual + LN (MHA) ----------
    if (tid < 62) {
      int r = tid;
      float o[10], xr[10], e8[8], v[10];
      for (int d = 0; d < DM; ++d) { o[d] = O1[r * 16 + d]; xr[d] = Xp[r * 16 + d]; }
      for (int j = 0; j < 8; ++j) {
        float s = L.ab1[j];
        for (int d = 0; d < DM; ++d) s += o[d] * L.aw1[d * 8 + j];
        e8[j] = eluf(s);
      }
      float mean = 0.f;
      for (int d = 0; d < DM; ++d) {
        float s = L.ab2[d];
        for (int j = 0; j < 8; ++j) s += e8[j] * L.aw2[j * DM + d];
        v[d] = o[d] + s + xr[d];
        mean += v[d];
      }
      mean /= (float)DM;
      float var = 0.f;
      for (int d = 0; d < DM; ++d) { float t = v[d] - mean; var += t * t; }
      var /= (float)DM;
      float inv = rsqrtf(var + 1e-5f);
      for (int d = 0; d < DM; ++d) Xp[r * 16 + d] = (v[d] - mean) * inv;
    }
    __syncthreads();

    // ---------- FFN + adapter + residual + LN ----------
    if (tid < 62) {
      int r = tid;
      float xr[10], u[10], o[10], e8[8], v[10];
      for (int d = 0; d < DM; ++d) xr[d] = Xp[r * 16 + d];
      for (int j = 0; j < 10; ++j) {
        float s = 0.f;
        for (int d = 0; d < DM; ++d) s += xr[d] * L.f1[d * 10 + j];
        u[j] = fmaxf(s, 0.f);
      }
      for (int d = 0; d < DM; ++d) {
        float s = 0.f;
        for (int j = 0; j < 10; ++j) s += u[j] * L.f2[j * DM + d];
        o[d] = s;
      }
      for (int j = 0; j < 8; ++j) {
        float s = L.fb1[j];
        for (int d = 0; d < DM; ++d) s += o[d] * L.fw1[d * 8 + j];
        e8[j] = eluf(s);
      }
      float mean = 0.f;
      for (int d = 0; d < DM; ++d) {
        float s = L.fb2[d];
        for (int j = 0; j < 8; ++j) s += e8[j] * L.fw2[j * DM + d];
        v[d] = o[d] + s + xr[d];
        mean += v[d];
      }
      mean /= (float)DM;
      float var = 0.f;
      for (int d = 0; d < DM; ++d) { float t = v[d] - mean; var += t * t; }
      var /= (float)DM;
      float inv = rsqrtf(var + 1e-5f);
      for (int d = 0; d < DM; ++d) Xp[r * 16 + d] = (v[d] - mean) * inv;
    }
    __syncthreads();
  }

  for (int i = tid; i < 62 * DM; i += 128) {
    int c = i / DM, d = i % DM;
    out[(size_t)(b * 62 + c) * out_ld + out_off + d] = Xp[c * 16 + d];
  }
}

// ============================================================
// feat = cat(4096x620) @ lin_w(620x64) + lin_b ; out = feat @ lin2_w + lin2_b
// WMMA f32: each wave one 16-row M-tile, 4 N-tiles, K=620 (155 steps of 4)
// ============================================================
__global__ __launch_bounds__(128)
void k_final(const float* __restrict__ fin, const float* __restrict__ lw,
             const float* __restrict__ lb, const float* __restrict__ l2w,
             const float* __restrict__ l2b, float* __restrict__ outO,
             float* __restrict__ outF)
{
  __shared__ float shF[64 * 64];
  const int tid = threadIdx.x;
  const int wave = tid >> 5, lane = tid & 31;
  const int lh = lane & 15, hi = lane >> 4;
  const int rowbase = blockIdx.x * 64;
  const int mb = rowbase + wave * 16;

  v8f a0 = {0.f, 0.f, 0.f, 0.f, 0.f, 0.f, 0.f, 0.f};
  v8f a1 = a0, a2 = a0, a3 = a0;
  for (int ks = 0; ks < 155; ++ks) {
    int k0 = ks * 4 + hi * 2;
    v2f a, b0, b1, b2, b3;
    a.x = fin[(size_t)(mb + lh) * 620 + k0];
    a.y = fin[(size_t)(mb + lh) * 620 + k0 + 1];
    b0.x = lw[k0 * 64 + lh];        b0.y = lw[(k0 + 1) * 64 + lh];
    b1.x = lw[k0 * 64 + 16 + lh];   b1.y = lw[(k0 + 1) * 64 + 16 + lh];
    b2.x = lw[k0 * 64 + 32 + lh];   b2.y = lw[(k0 + 1) * 64 + 32 + lh];
    b3.x = lw[k0 * 64 + 48 + lh];   b3.y = lw[(k0 + 1) * 64 + 48 + lh];
    a0 = wmma4(a, b0, a0);
    a1 = wmma4(a, b1, a1);
    a2 = wmma4(a, b2, a2);
    a3 = wmma4(a, b3, a3);
  }
  for (int r = 0; r < 8; ++r) {
    int row = wave * 16 + r + hi * 8;
    shF[row * 64 + lh]      = a0[r] + lb[lh];
    shF[row * 64 + 16 + lh] = a1[r] + lb[16 + lh];
    shF[row * 64 + 32 + lh] = a2[r] + lb[32 + lh];
    shF[row * 64 + 48 + lh] = a3[r] + lb[48 + lh];
  }
  __syncthreads();
  for (int i = tid; i < 64 * 64; i += 128)
    outF[(size_t)(rowbase + i / 64) * 64 + (i & 63)] = shF[i];
  for (int i = tid; i < 64 * 3; i += 128) {
    int r = i / 3, j = i % 3;
    float s = l2b[j];
    for (int c = 0; c < 64; ++c) s += shF[r * 64 + c] * l2w[c * 3 + j];
    outO[(size_t)(rowbase + r) * 3 + j] = s;
  }
}

// ============================================================
// host side
// ============================================================
static LayerP mk_layer(void* const* d_in, int base) {
  LayerP p;
  p.wq  = (const float*)d_in[base + 0];
  p.wk  = (const float*)d_in[base + 1];
  p.wv  = (const float*)d_in[base + 2];
  p.fc  = (const float*)d_in[base + 3];
  p.aw1 = (const float*)d_in[base + 4];
  p.ab1 = (const float*)d_in[base + 5];
  p.aw2 = (const float*)d_in[base + 6];
  p.ab2 = (const float*)d_in[base + 7];
  p.f1  = (const float*)d_in[base + 8];
  p.f2  = (const float*)d_in[base + 9];
  p.fw1 = (const float*)d_in[base + 10];
  p.fb1 = (const float*)d_in[base + 11];
  p.fw2 = (const float*)d_in[base + 12];
  p.fb2 = (const float*)d_in[base + 13];
  return p;
}

extern "C" void kernel_launch(void* const* d_in, const int* in_sizes, int n_in,
                              void* d_out, int out_size, void* d_ws, size_t ws_size,
                              hipStream_t stream)
{
  (void)in_sizes; (void)n_in; (void)out_size; (void)ws_size;
  float* ws = (float*)d_ws;

  const float* x   = (const float*)d_in[0];
  const float* A   = (const float*)d_in[1];
  const float* gw1 = (const float*)d_in[2];
  const float* gw2 = (const float*)d_in[3];
  // gcn1 leaves 4..9 : k1,g1,b1,k2,g2,b2 ; gcn2 leaves 10..15
  const float* g1k1 = (const float*)d_in[4];
  const float* g1g1 = (const float*)d_in[5];
  const float* g1b1 = (const float*)d_in[6];
  const float* g1k2 = (const float*)d_in[7];
  const float* g1g2 = (const float*)d_in[8];
  const float* g1b2 = (const float*)d_in[9];
  const float* g2k1 = (const float*)d_in[10];
  const float* g2g1 = (const float*)d_in[11];
  const float* g2b1 = (const float*)d_in[12];
  const float* g2k2 = (const float*)d_in[13];
  const float* g2g2 = (const float*)d_in[14];
  const float* g2b2 = (const float*)d_in[15];

  float* Ads  = ws + OFF_ADS;
  float* Lm   = ws + OFF_L;
  float* FIN1 = ws + OFF_FIN1;
  float* FIN2 = ws + OFF_FIN2;
  float* P1   = ws + OFF_P1;
  float* P2   = ws + OFF_P2;
  float* H1   = ws + OFF_H1;
  float* H2   = ws + OFF_H2;
  float* F12  = ws + OFF_F12;
  float* CAT  = ws + OFF_CAT;

  k_prep<<<1, 256, 0, stream>>>(A, gw1, gw2, Ads, Lm);
  k_conv1<<<dim3(BATCH, 2), 128, 0, stream>>>(x, g1k1, g2k1, H1, P1);
  k_finstats<<<1, 256, 0, stream>>>(P1, FIN1, g1g1, g1b1, g2g1, g2b1,
                                    1.f / (4096.f * 60.f));
  k_conv2<<<dim3(BATCH, 2), 128, 0, stream>>>(H1, FIN1, g1k2, g2k2, H2, P2);
  k_finstats<<<1, 256, 0, stream>>>(P2, FIN2, g1g2, g1b2, g2g2, g2b2,
                                    1.f / (4096.f * 62.f));
  k_gcn_out<<<dim3(BATCH, 2), 128, 0, stream>>>(x, H2, FIN2, Lm, F12);

  EncP e1, e2, e0;
  e1.L[0] = mk_layer(d_in, 16); e1.L[1] = mk_layer(d_in, 30);
  e2.L[0] = mk_layer(d_in, 44); e2.L[1] = mk_layer(d_in, 58);
  e0.L[0] = mk_layer(d_in, 72); e0.L[1] = mk_layer(d_in, 86);

  k_encoder<5><<<BATCH, 128, 0, stream>>>(e1, F12, 5, 0, CAT, 10, 0, Ads);
  k_encoder<5><<<BATCH, 128, 0, stream>>>(e2, F12 + (size_t)BATCH * 310, 5, 0,
                                          CAT, 10, 5, Ads);
  k_encoder<10><<<BATCH, 128, 0, stream>>>(e0, CAT, 10, 0, CAT, 10, 0, Ads);

  float* outO = (float*)d_out;
  float* outF = outO + (size_t)BATCH * 3;
  k_final<<<64, 128, 0, stream>>>(CAT, (const float*)d_in[100],
                                  (const float*)d_in[101],
                                  (const float*)d_in[102],
                                  (const float*)d_in[103], outO, outF);
}